// SGC_17428977287560
// MI455X (gfx1250) — compile-verified
//
#include <hip/hip_runtime.h>
#include <hip/hip_bf16.h>

#define N_NODES   100000
#define N_EDGES   1600000
#define IN_FEATS  256
#define N_HIDDEN  128
#define N_CLASSES 40

typedef float v2f __attribute__((ext_vector_type(2)));
typedef float v8f __attribute__((ext_vector_type(8)));

// ---------------- degree / norm ----------------
__global__ void sgc_deg_kernel(const int* __restrict__ dst, float* __restrict__ deg) {
    int e = blockIdx.x * blockDim.x + threadIdx.x;
    if (e < N_EDGES) unsafeAtomicAdd(&deg[dst[e]], 1.0f);
}

__global__ void sgc_norm_kernel(float* __restrict__ deg_norm) {
    int i = blockIdx.x * blockDim.x + threadIdx.x;
    if (i < N_NODES) {
        float d = deg_norm[i];
        d = d < 1.0f ? 1.0f : d;
        deg_norm[i] = 1.0f / sqrtf(d);
    }
}

// ---------------- one propagation hop: y[dst] += norm[src]*norm[dst] * x[src] ----------------
// 64 threads per edge, float4 per thread (64*4 = 256 feats)
__global__ void sgc_spmm_scatter(const float* __restrict__ x,
                                 const int* __restrict__ src,
                                 const int* __restrict__ dst,
                                 const float* __restrict__ norm,
                                 float* __restrict__ y) {
    int t = blockIdx.x * blockDim.x + threadIdx.x;
    int e = t >> 6;
    if (e >= N_EDGES) return;
    int f = (t & 63) << 2;
    int s = src[e];
    int d = dst[e];
    float w = norm[s] * norm[d];
    const float4 v = *reinterpret_cast<const float4*>(x + (size_t)s * IN_FEATS + f);
    float* out = y + (size_t)d * IN_FEATS + f;
    unsafeAtomicAdd(out + 0, v.x * w);
    unsafeAtomicAdd(out + 1, v.y * w);
    unsafeAtomicAdd(out + 2, v.z * w);
    unsafeAtomicAdd(out + 3, v.w * w);
}

// ---------------- f32 WMMA GEMM: C[M x NC] = A[M x K] @ B[K x NC] + bias, optional ReLU -----
// One wave (32 threads) computes one 16x16 output tile via V_WMMA_F32_16X16X4_F32.
// blockIdx.x = row tile, blockIdx.y = col tile. NC = valid cols = ldb = ldc.
// GUARD=0: grid covers exactly NC columns (no edge handling at all).
// GUARD=1: last col tile is partial. B loads are clamped in-bounds instead of
//          zero-filled: lane l's B values feed ONLY output column (l&15) of the
//          tile, so garbage from clamped loads lands only in columns we never
//          store. This keeps the inner loop free of EXEC manipulation.
template <int K, int NC, int RELU, int GUARD>
__global__ __launch_bounds__(32)
void sgc_wmma_gemm(const float* __restrict__ A,
                   const float* __restrict__ B,
                   const float* __restrict__ bias,
                   float* __restrict__ C) {
    const int lane = threadIdx.x;        // 0..31
    const int half = lane >> 4;          // 0 or 1
    const int l15  = lane & 15;
    const int row  = blockIdx.x * 16 + l15;   // A row this lane loads
    const int col  = blockIdx.y * 16 + l15;   // B/C column this lane owns
    const int bcol = GUARD ? (col < NC ? col : NC - 1) : col;

    const float* arow = A + (size_t)row * K;
    const float* bcolp = B + bcol;
    v8f c = {};
#pragma unroll 4
    for (int k0 = 0; k0 < K; k0 += 4) {
        const int ka = k0 + half * 2;
        v2f a, b;
        a.x = arow[ka];
        a.y = arow[ka + 1];
        b.x = bcolp[(size_t)(ka)     * NC];
        b.y = bcolp[(size_t)(ka + 1) * NC];
        c = __builtin_amdgcn_wmma_f32_16x16x4_f32(
                /*neg_a=*/false, a, /*neg_b=*/false, b,
                /*c_mod=*/(short)0, c, /*reuse_a=*/false, /*reuse_b=*/false);
    }

    if (!GUARD || col < NC) {
        const float bv = bias[col];
        const int rbase = blockIdx.x * 16 + half * 8;
#pragma unroll
        for (int v = 0; v < 8; ++v) {
            float o = c[v] + bv;
            if (RELU) o = fmaxf(o, 0.0f);
            C[(size_t)(rbase + v) * NC + col] = o;
        }
    }
}

extern "C" void kernel_launch(void* const* d_in, const int* in_sizes, int n_in,
                              void* d_out, int out_size, void* d_ws, size_t ws_size,
                              hipStream_t stream) {
    const float* features = (const float*)d_in[0];
    const int*   src      = (const int*)d_in[1];
    const int*   dst      = (const int*)d_in[2];
    const float* W_sgc    = (const float*)d_in[3];
    const float* b_sgc    = (const float*)d_in[4];
    const float* W_lin    = (const float*)d_in[5];
    const float* b_lin    = (const float*)d_in[6];
    float*       out      = (float*)d_out;

    char* ws = (char*)d_ws;
    size_t off = 0;
    float* norm = (float*)(ws + off); off += (((size_t)N_NODES * 4) + 255) & ~(size_t)255;
    float* bufA = (float*)(ws + off); off += (size_t)N_NODES * IN_FEATS * 4;
    float* bufB = (float*)(ws + off); off += (size_t)N_NODES * IN_FEATS * 4;

    // 1) degree + norm
    hipMemsetAsync(norm, 0, (size_t)N_NODES * 4, stream);
    sgc_deg_kernel<<<(N_EDGES + 255) / 256, 256, 0, stream>>>(dst, norm);
    sgc_norm_kernel<<<(N_NODES + 255) / 256, 256, 0, stream>>>(norm);

    // 2) hop 1: features -> bufA
    hipMemsetAsync(bufA, 0, (size_t)N_NODES * IN_FEATS * 4, stream);
    {
        long long threads = (long long)N_EDGES * 64;
        int blocks = (int)((threads + 255) / 256);
        sgc_spmm_scatter<<<blocks, 256, 0, stream>>>(features, src, dst, norm, bufA);
    }

    // 3) hop 2: bufA -> bufB
    hipMemsetAsync(bufB, 0, (size_t)N_NODES * IN_FEATS * 4, stream);
    {
        long long threads = (long long)N_EDGES * 64;
        int blocks = (int)((threads + 255) / 256);
        sgc_spmm_scatter<<<blocks, 256, 0, stream>>>(bufA, src, dst, norm, bufB);
    }

    // 4) GEMM1 + bias + ReLU: bufB (N x 256) @ W_sgc (256 x 128) -> bufA (N x 128)
    sgc_wmma_gemm<IN_FEATS, N_HIDDEN, 1, 0>
        <<<dim3(N_NODES / 16, N_HIDDEN / 16), 32, 0, stream>>>(bufB, W_sgc, b_sgc, bufA);

    // 5) GEMM2 + bias: bufA (N x 128) @ W_lin (128 x 40) -> out (N x 40)
    sgc_wmma_gemm<N_HIDDEN, N_CLASSES, 0, 1>
        <<<dim3(N_NODES / 16, (N_CLASSES + 15) / 16), 32, 0, stream>>>(bufA, W_lin, b_lin, out);
}